// AttenuationToRainRate_41893111005595
// MI455X (gfx1250) — compile-verified
//
#include <hip/hip_runtime.h>

#define B_SZ 256
#define T_SZ 8192

typedef float v2f __attribute__((ext_vector_type(2)));
typedef float v8f __attribute__((ext_vector_type(8)));

// ---------------------------------------------------------------------------
// Style MLP GEMM: out(M x N) = act(A(M x K) @ W(K x N) + bias)
// One wave (32 lanes) per 16x16 output tile, V_WMMA_F32_16X16X4_F32 chained
// over K.  Layouts per CDNA5 ISA 7.12.2:
//   A 16x4 : lane = M row (0-15 in both halves), VGPR0 = K {0 lo | 2 hi},
//            VGPR1 = K {1 lo | 3 hi}
//   B 4x16 : lane = N col, same K striping as A
//   C/D    : VGPR r = row {r lo | r+8 hi}, lane = N col
// ---------------------------------------------------------------------------
template <int K, bool RELU>
__global__ __launch_bounds__(32) void style_gemm(const float* __restrict__ A,
                                                 const float* __restrict__ W,
                                                 const float* __restrict__ bias,
                                                 float* __restrict__ out, int N) {
  const int lane = threadIdx.x;
  const int col  = lane & 15;
  const int hi   = lane >> 4;   // 0: lanes 0-15, 1: lanes 16-31
  const int kk   = hi * 2;
  const int n0   = blockIdx.x * 16;
  const int m0   = blockIdx.y * 16;
  const int row  = m0 + col;

  const float bv = bias[n0 + col];
  v8f c;
#pragma unroll
  for (int r = 0; r < 8; ++r) c[r] = bv;

#pragma unroll
  for (int k = 0; k < K; k += 4) {
    v2f a, b;
    a.x = A[row * K + k + kk];
    a.y = A[row * K + k + kk + 1];
    b.x = W[(k + kk) * N + n0 + col];
    b.y = W[(k + kk + 1) * N + n0 + col];
    // (neg_a, A, neg_b, B, c_mod, C, reuse_a, reuse_b)
    c = __builtin_amdgcn_wmma_f32_16x16x4_f32(false, a, false, b, (short)0, c,
                                              false, false);
  }

#pragma unroll
  for (int r = 0; r < 8; ++r) {
    float v = c[r];
    if (RELU) v = fmaxf(v, 0.0f);
    out[(m0 + r + hi * 8) * N + n0 + col] = v;
  }
}

// ---------------------------------------------------------------------------
// Main fused per-timestep MLP + AdaIN + leaky ReLU.  Lane-per-timestep:
// the 8 channels of one timestep live in 8 VGPRs of one lane, so the AdaIN
// reduction (mean/std over channels, ddof=1) is entirely in-lane.
// Layer-major schedule: all ITER timesteps advance one layer at a time so
// only ONE 8x8 weight matrix (64 VGPRs) is live at once (~180 VGPR peak,
// no spills), instead of all three (~320 VGPRs).
// ---------------------------------------------------------------------------
__device__ __forceinline__ void adain_lrelu(float h[8], const float sc[8],
                                            const float bi[8]) {
  float sum = ((h[0] + h[1]) + (h[2] + h[3])) + ((h[4] + h[5]) + (h[6] + h[7]));
  float mu  = sum * 0.125f;
  float d[8];
  float ss = 0.0f;
#pragma unroll
  for (int j = 0; j < 8; ++j) {
    d[j] = h[j] - mu;
    ss   = fmaf(d[j], d[j], ss);
  }
  float sigma = __builtin_amdgcn_sqrtf(ss * (1.0f / 7.0f)) + 1e-6f;  // ddof=1
  float inv   = __builtin_amdgcn_rcpf(sigma);
#pragma unroll
  for (int j = 0; j < 8; ++j) {
    float v = fmaf(d[j] * inv, sc[j], bi[j]);
    h[j]    = fmaxf(v, 0.01f * v);  // leaky relu
  }
}

template <int ITER>
__global__ __launch_bounds__(256) void fused_pointwise_mlp(
    const float* __restrict__ x, const float* __restrict__ style,
    const float* __restrict__ w1, const float* __restrict__ b1,
    const float* __restrict__ w2, const float* __restrict__ b2,
    const float* __restrict__ w3, const float* __restrict__ b3,
    const float* __restrict__ w4, const float* __restrict__ b4,
    const float* __restrict__ w5, const float* __restrict__ b5,
    float* __restrict__ out) {
  const int b     = blockIdx.y;
  const long base = (long)b * T_SZ + (long)blockIdx.x * (256 * ITER) + threadIdx.x;
  const float* st = style + b * 64;  // AdaIN params: st[16*l + 2*c + {0,1}]

  float H[ITER][8];  // activations: [timestep][channel], register-resident

  // ---- Layer 1 (1 -> 8) + AdaIN(s1) + lrelu -------------------------------
  {
    float w1v[8], b1v[8], sc[8], bi[8];
#pragma unroll
    for (int j = 0; j < 8; ++j) {
      w1v[j] = w1[j];
      b1v[j] = b1[j];
      sc[j]  = st[2 * j];
      bi[j]  = st[2 * j + 1];
    }
#pragma unroll
    for (int i = 0; i < ITER; ++i) {
      const float xv = x[base + i * 256];
#pragma unroll
      for (int j = 0; j < 8; ++j) H[i][j] = fmaf(xv, w1v[j], b1v[j]);
      adain_lrelu(H[i], sc, bi);
    }
  }

  // ---- Layers 2..4 (8 -> 8) + AdaIN + lrelu, one weight matrix at a time --
  auto mid_layer = [&](const float* __restrict__ w,
                       const float* __restrict__ bias, int l) {
    float W[64];
#pragma unroll
    for (int i = 0; i < 16; ++i)
      reinterpret_cast<float4*>(W)[i] = reinterpret_cast<const float4*>(w)[i];
    float bv[8], sc[8], bi[8];
#pragma unroll
    for (int j = 0; j < 8; ++j) {
      bv[j] = bias[j];
      sc[j] = st[l * 16 + 2 * j];
      bi[j] = st[l * 16 + 2 * j + 1];
    }
#pragma unroll
    for (int i = 0; i < ITER; ++i) {
      float g[8];
#pragma unroll
      for (int j = 0; j < 8; ++j) g[j] = bv[j];
#pragma unroll
      for (int k = 0; k < 8; ++k)
#pragma unroll
        for (int j = 0; j < 8; ++j) g[j] = fmaf(H[i][k], W[k * 8 + j], g[j]);
      adain_lrelu(g, sc, bi);
#pragma unroll
      for (int j = 0; j < 8; ++j) H[i][j] = g[j];  // coalesced by regalloc
    }
  };
  mid_layer(w2, b2, 1);
  mid_layer(w3, b3, 2);
  mid_layer(w4, b4, 3);

  // ---- Layer 5 (8 -> 1) + lrelu + store -----------------------------------
  {
    float w5v[8];
#pragma unroll
    for (int j = 0; j < 8; ++j) w5v[j] = w5[j];
    const float b5s = b5[0];
#pragma unroll
    for (int i = 0; i < ITER; ++i) {
      float o = b5s;
#pragma unroll
      for (int k = 0; k < 8; ++k) o = fmaf(H[i][k], w5v[k], o);
      out[base + i * 256] = fmaxf(o, 0.01f * o);
    }
  }
}

// ---------------------------------------------------------------------------
extern "C" void kernel_launch(void* const* d_in, const int* in_sizes, int n_in,
                              void* d_out, int out_size, void* d_ws,
                              size_t ws_size, hipStream_t stream) {
  const float* x   = (const float*)d_in[0];
  const float* md  = (const float*)d_in[1];
  const float* mw1 = (const float*)d_in[2];
  const float* mb1 = (const float*)d_in[3];
  const float* mw2 = (const float*)d_in[4];
  const float* mb2 = (const float*)d_in[5];
  const float* mw3 = (const float*)d_in[6];
  const float* mb3 = (const float*)d_in[7];
  const float* w1  = (const float*)d_in[8];
  const float* b1  = (const float*)d_in[9];
  const float* w2  = (const float*)d_in[10];
  const float* b2  = (const float*)d_in[11];
  const float* w3  = (const float*)d_in[12];
  const float* b3  = (const float*)d_in[13];
  const float* w4  = (const float*)d_in[14];
  const float* b4  = (const float*)d_in[15];
  const float* w5  = (const float*)d_in[16];
  const float* b5  = (const float*)d_in[17];
  float* out = (float*)d_out;

  // Workspace: style activations (all fp32)
  float* s1 = (float*)d_ws;       // 256 x 64
  float* s2 = s1 + 256 * 64;      // 256 x 128
  float* s3 = s2 + 256 * 128;     // 256 x 64  (final style: scale/bias pairs)

  // Style MLP via WMMA f32 16x16x4 (one wave per 16x16 tile).
  style_gemm<16, true><<<dim3(64 / 16, B_SZ / 16), 32, 0, stream>>>(md, mw1, mb1, s1, 64);
  style_gemm<64, true><<<dim3(128 / 16, B_SZ / 16), 32, 0, stream>>>(s1, mw2, mb2, s2, 128);
  style_gemm<128, false><<<dim3(64 / 16, B_SZ / 16), 32, 0, stream>>>(s2, mw3, mb3, s3, 64);

  // Main fused per-timestep MLP (compute-bound, fp32 VALU, lane-per-timestep).
  constexpr int ITER = 8;
  fused_pointwise_mlp<ITER><<<dim3(T_SZ / (256 * ITER), B_SZ), 256, 0, stream>>>(
      x, s3, w1, b1, w2, b2, w3, b3, w4, b4, w5, b5, out);
}